// RGCN_30743375905152
// MI455X (gfx1250) — compile-verified
//
#include <hip/hip_runtime.h>
#include <hip/hip_bf16.h>

typedef float v2f __attribute__((ext_vector_type(2)));
typedef float v8f __attribute__((ext_vector_type(8)));

#define HD 128   // H_DIM == OUT_DIM == 128

// Native fp32 global atomic add (no-return form), device scope so the RMW is
// performed at L2 and coherent across all WGPs.
__device__ __forceinline__ void gatomic_fadd(float* p, float v) {
    asm volatile("global_atomic_add_f32 %0, %1, off scope:SCOPE_DEV"
                 :: "v"(p), "v"(v) : "memory");
}

// ---------------------------------------------------------------------------
// Zero-fill d_out (scatter accumulates into it).
// ---------------------------------------------------------------------------
__global__ __launch_bounds__(256) void rgcn_zero_kernel(float4* __restrict__ p, int n4) {
    int i = blockIdx.x * blockDim.x + threadIdx.x;
    if (i < n4) p[i] = make_float4(0.f, 0.f, 0.f, 0.f);
}

// ---------------------------------------------------------------------------
// GEMM: hrel[n, :] = emb[node_ids[n], :] @ Wr   (one relation slice)
// Block = 256 threads = 8 waves.  Block owns 128 rows; wave owns 16 rows.
// Wr (128x128, 64KB) staged in LDS; A tile kept in registers in WMMA layout.
// Uses V_WMMA_F32_16X16X4_F32, K stepped 4 at a time (32 steps).
// ---------------------------------------------------------------------------
__global__ __launch_bounds__(256) void rgcn_gemm_kernel(
    const float* __restrict__ emb,    // [N, 128]
    const int*   __restrict__ nid,    // [N]
    const float* __restrict__ Wr,     // [128, 128] (relation slice)
    float*       __restrict__ hrel,   // [N, 128]
    int n_nodes)
{
    __shared__ float sB[HD * HD];     // 64 KB

    const int t = threadIdx.x;

    // Cooperative load of W[r] into LDS (float4, fully coalesced).
    {
        const float4* Wv = (const float4*)Wr;
        float4*       sv = (float4*)sB;
        #pragma unroll
        for (int i = 0; i < 16; ++i)
            sv[t + 256 * i] = Wv[t + 256 * i];
    }
    __syncthreads();

    const int wave = t >> 5;
    const int lane = t & 31;
    const int mrow = lane & 15;            // M row within 16x16 tile
    const int koff = (lane >> 4) << 1;     // lanes 16-31 hold K+2 (ISA A layout)
    const int row  = blockIdx.x * 128 + wave * 16 + mrow;

    // Preload A fragments: a[j] = A[M=mrow][K=4j+koff .. 4j+koff+1]
    v2f a[32];
    if (row < n_nodes) {
        const float* hp = emb + (size_t)nid[row] * HD + koff;
        #pragma unroll
        for (int j = 0; j < 32; ++j)
            a[j] = *(const v2f*)(hp + 4 * j);
    } else {
        #pragma unroll
        for (int j = 0; j < 32; ++j) { v2f z = {}; a[j] = z; }
    }

    const int mbase = blockIdx.x * 128 + wave * 16 + ((lane >> 4) << 3);

    #pragma unroll
    for (int nt = 0; nt < 8; ++nt) {
        const int ncol = nt * 16 + mrow;   // this lane's N column of B/C
        v8f acc = {};
        #pragma unroll
        for (int j = 0; j < 32; ++j) {
            // B layout mirrors A: VGPR v holds K row (4j+v) in lanes 0-15,
            // K row (4j+v+2) in lanes 16-31; N striped across lanes.
            v2f b;
            b.x = sB[(4 * j + koff    ) * HD + ncol];
            b.y = sB[(4 * j + koff + 1) * HD + ncol];
            acc = __builtin_amdgcn_wmma_f32_16x16x4_f32(
                      false, a[j], false, b, (short)0, acc, false, false);
        }
        // C/D layout: VGPR v -> M = v (lanes 0-15) or v+8 (lanes 16-31), N = lane&15
        #pragma unroll
        for (int v = 0; v < 8; ++v) {
            const int rr = mbase + v;
            if (rr < n_nodes) hrel[(size_t)rr * HD + ncol] = acc[v];
        }
    }
}

// ---------------------------------------------------------------------------
// Scatter: for edges with rel[e]==r:  out[dst[e], :] += hrel[src[e], :] * norm[e]
// One wave (32 lanes) per edge; each lane owns a float4 slice of the 128-wide
// row -> one coalesced 512B gather per edge, 4 native f32 atomics per lane.
// ---------------------------------------------------------------------------
__global__ __launch_bounds__(256) void rgcn_scatter_kernel(
    const int*   __restrict__ src,
    const int*   __restrict__ dst,
    const int*   __restrict__ rel,
    const float* __restrict__ norm,
    const float* __restrict__ hrel,
    float*       __restrict__ out,
    int n_edges, int r)
{
    const int e = (int)((blockIdx.x * blockDim.x + threadIdx.x) >> 5);
    if (e >= n_edges) return;
    if (rel[e] != r) return;            // wave-uniform predicate

    const int   lane = threadIdx.x & 31;
    const float nm   = norm[e];
    const float4 v   = *(const float4*)(hrel + (size_t)src[e] * HD + lane * 4);
    float* op        = out + (size_t)dst[e] * HD + lane * 4;

    gatomic_fadd(op + 0, v.x * nm);
    gatomic_fadd(op + 1, v.y * nm);
    gatomic_fadd(op + 2, v.z * nm);
    gatomic_fadd(op + 3, v.w * nm);
}

// ---------------------------------------------------------------------------
extern "C" void kernel_launch(void* const* d_in, const int* in_sizes, int n_in,
                              void* d_out, int out_size, void* d_ws, size_t ws_size,
                              hipStream_t stream) {
    const int*   node_ids = (const int*)  d_in[0];
    const int*   src      = (const int*)  d_in[1];
    const int*   dst      = (const int*)  d_in[2];
    const int*   rel      = (const int*)  d_in[3];
    const float* norm     = (const float*)d_in[4];
    const float* emb      = (const float*)d_in[5];
    const float* W        = (const float*)d_in[6];
    float*       out      = (float*)      d_out;
    float*       hrel     = (float*)      d_ws;     // [n_nodes, 128] scratch

    const int n_nodes = in_sizes[0];
    const int n_edges = in_sizes[1];
    const int n_rels  = in_sizes[6] / (HD * HD);

    // 1) zero output
    const int n4 = out_size / 4;
    rgcn_zero_kernel<<<(n4 + 255) / 256, 256, 0, stream>>>((float4*)out, n4);

    // 2) per relation: fp32 WMMA GEMM into hrel, then edge scatter-accumulate
    const int mblocks = (n_nodes + 127) / 128;
    const int sblocks = (n_edges + 7) / 8;          // 8 waves (edges) per block
    for (int r = 0; r < n_rels; ++r) {
        rgcn_gemm_kernel<<<mblocks, 256, 0, stream>>>(
            emb, node_ids, W + (size_t)r * HD * HD, hrel, n_nodes);
        rgcn_scatter_kernel<<<sblocks, 256, 0, stream>>>(
            src, dst, rel, norm, hrel, out, n_edges, r);
    }
}